// ComputeFFTDelta_18743237279903
// MI455X (gfx1250) — compile-verified
//
#include <hip/hip_runtime.h>
#include <cmath>

// ---------------------------------------------------------------------------
// Fourier-accountant delta on MI455X (gfx1250, wave32).
//  K1: fused 16M-element stream: logf/expf, fp32 global-atomic histograms,
//      f64-atomic logsumexp sums (alpha_plus / alpha_minus shared by both calls)
//  K2: four-step FFT stage 1: 512-pt column FFTs built from wave32 256-pt FFTs,
//      each 256-pt FFT = two 16x16 complex matmuls on V_WMMA_F32_16X16X4_F32
//  K3: stage 2 (row FFT-256 via WMMA) fused with analytic W_f tail reduction
//  K4: scalar finalize (Theorem-10 error terms, dist_events composition)
// fftshifts are folded into (-1)^f phases; F^4 = P^4 since M=4 is even.
// ---------------------------------------------------------------------------

#define FFTN 131072
#define TWO_PI_F 6.28318530717958647692f
#define TWO_PI_D 6.28318530717958647692

typedef __attribute__((ext_vector_type(2))) float v2f;
typedef __attribute__((ext_vector_type(8))) float v8f;

__device__ __forceinline__ v8f wmma4(v2f a, v2f b, v8f c) {
  // D(16x16,f32) = A(16x4,f32) x B(4x16,f32) + C
  return __builtin_amdgcn_wmma_f32_16x16x4_f32(false, a, false, b, (short)0, c,
                                               false, false);
}

// ---------------------------------------------------------------------------
// 256-point complex DFT executed by ONE wave32 using WMMA.
// in/out: 256-entry re/im arrays in LDS. mid: 256-entry scratch (re/im).
// X[k + 16 q] = sum_t in[t] w256^{t(k+16q)},  w = e^{-2pi i /256}.
// Decompose t = r + 16 c:  B = M0 x W16 ; twiddle w256^{r k} ; C = W16 x B'.
// ---------------------------------------------------------------------------
__device__ static inline void fft256_wave(const float* inRe, const float* inIm,
                                          float* midRe, float* midIm,
                                          float* outRe, float* outIm, int lane) {
  const int col = lane & 15;   // A-operand M / B-operand N / C-D column
  const int half = lane >> 4;  // lane group select

  // ---------------- Stage 1: B = M0 * W16  (data is A operand) -------------
  v2f aRe[4], aIm[4], bWr[4], bWi[4], bWiN[4];
#pragma unroll
  for (int t = 0; t < 4; ++t) {
    const int c0 = 4 * t + 2 * half;  // K for vgpr 0
    const int c1 = c0 + 1;            // K for vgpr 1
    // A operand (16x4 slice of M0): element (M=col, K=c) = x[col + 16 c]
    v2f ar, ai;
    ar.x = inRe[col + 16 * c0]; ar.y = inRe[col + 16 * c1];
    ai.x = inIm[col + 16 * c0]; ai.y = inIm[col + 16 * c1];
    aRe[t] = ar; aIm[t] = ai;
    // B operand (4x16 slice of W16): element (K=c, N=col) = w16^{c*col}
    float s0, cc0, s1, cc1;
    sincosf(TWO_PI_F * (float)((c0 * col) & 15) * (1.0f / 16.0f), &s0, &cc0);
    sincosf(TWO_PI_F * (float)((c1 * col) & 15) * (1.0f / 16.0f), &s1, &cc1);
    v2f wr, wi, win;
    wr.x = cc0;  wr.y = cc1;     // Re(w16) = cos
    wi.x = -s0;  wi.y = -s1;     // Im(w16) = -sin
    win.x = s0;  win.y = s1;     // -Im(w16)
    bWr[t] = wr; bWi[t] = wi; bWiN[t] = win;
  }
  v8f accR = {}, accI = {};
#pragma unroll
  for (int t = 0; t < 4; ++t) accR = wmma4(aRe[t], bWr[t], accR);
#pragma unroll
  for (int t = 0; t < 4; ++t) accR = wmma4(aIm[t], bWiN[t], accR);  // -Ai*Wi
#pragma unroll
  for (int t = 0; t < 4; ++t) accI = wmma4(aRe[t], bWi[t], accI);
#pragma unroll
  for (int t = 0; t < 4; ++t) accI = wmma4(aIm[t], bWr[t], accI);

  // Twiddle w256^{r' k} in C/D layout, spill to LDS for operand re-layout.
#pragma unroll
  for (int v = 0; v < 8; ++v) {
    const int rp = v + 8 * half;  // row r'
    float s, c;
    sincosf(TWO_PI_F * (float)((rp * col) & 255) * (1.0f / 256.0f), &s, &c);
    const float br = accR[v], bi = accI[v];
    midRe[rp * 16 + col] = br * c + bi * s;  // (br+i bi)(cos - i sin)
    midIm[rp * 16 + col] = bi * c - br * s;
  }
  __syncthreads();

  // ---------------- Stage 2: C = W16 * B'  (data is B operand) -------------
  v2f aWr[4], aWi[4], aWiN[4], bRe[4], bIm[4];
#pragma unroll
  for (int t = 0; t < 4; ++t) {
    const int k0 = 4 * t + 2 * half;  // K = r' for vgpr 0
    const int k1 = k0 + 1;
    // A operand (16x4 slice of W16): element (M=col, K=r') = w16^{col*r'}
    float s0, cc0, s1, cc1;
    sincosf(TWO_PI_F * (float)((col * k0) & 15) * (1.0f / 16.0f), &s0, &cc0);
    sincosf(TWO_PI_F * (float)((col * k1) & 15) * (1.0f / 16.0f), &s1, &cc1);
    v2f wr, wi, win;
    wr.x = cc0;  wr.y = cc1;
    wi.x = -s0;  wi.y = -s1;
    win.x = s0;  win.y = s1;
    aWr[t] = wr; aWi[t] = wi; aWiN[t] = win;
    // B operand (4x16 slice of B'): element (K=r', N=col)
    v2f br, bi;
    br.x = midRe[k0 * 16 + col]; br.y = midRe[k1 * 16 + col];
    bi.x = midIm[k0 * 16 + col]; bi.y = midIm[k1 * 16 + col];
    bRe[t] = br; bIm[t] = bi;
  }
  v8f cR = {}, cI = {};
#pragma unroll
  for (int t = 0; t < 4; ++t) cR = wmma4(aWr[t], bRe[t], cR);
#pragma unroll
  for (int t = 0; t < 4; ++t) cR = wmma4(aWiN[t], bIm[t], cR);  // -Wi*bi
#pragma unroll
  for (int t = 0; t < 4; ++t) cI = wmma4(aWr[t], bIm[t], cI);
#pragma unroll
  for (int t = 0; t < 4; ++t) cI = wmma4(aWi[t], bRe[t], cI);

#pragma unroll
  for (int v = 0; v < 8; ++v) {
    const int q = v + 8 * half;
    outRe[col + 16 * q] = cR[v];  // X[k2i + 16*k1i] = C[k1i][k2i]
    outIm[col + 16 * q] = cI[v];
  }
  __syncthreads();
}

// ---------------------------------------------------------------------------
// K0: zero hist + accumulators (d_ws is not re-poisoned between replays)
// ---------------------------------------------------------------------------
__global__ void k0_zero(float* __restrict__ hist, double* __restrict__ acc) {
  const int i = blockIdx.x * blockDim.x + threadIdx.x;
  if (i < 2 * FFTN) hist[i] = 0.0f;
  if (i < 4) acc[i] = 0.0;
}

// ---------------------------------------------------------------------------
// K1: fused stream pass (memory bound: 128 MB read -> ~5.5 us floor)
// ---------------------------------------------------------------------------
__global__ void k1_stream(const float* __restrict__ pA,
                          const float* __restrict__ pB,
                          float* __restrict__ histA, float* __restrict__ histB,
                          double* __restrict__ acc, float Lf, float DXf,
                          float LAMf, int n) {
  const int gid = blockIdx.x * blockDim.x + threadIdx.x;
  const int stride = gridDim.x * blockDim.x;
  float s1 = 0.0f, s2 = 0.0f;
  for (int i = gid; i < n; i += stride) {
    const float va = pA[i], vb = pB[i];
    const float la = logf(va), lb = logf(vb);
    const float pl = la - lb;
    // primal histogram: idx = ceil((L+pl)/dx), drop out-of-range (dump bucket)
    const int ia = (int)ceilf((Lf + pl) / DXf);
    if (ia >= 0 && ia < FFTN) atomicAdd(&histA[ia], va);
    // dual histogram: privacy loss negated, weights p_B
    const int ib = (int)ceilf((Lf - pl) / DXf);
    if (ib >= 0 && ib < FFTN) atomicAdd(&histB[ib], vb);
    // exp(alpha_plus) = sum exp(LAM*pl + logA); exp(alpha_minus) symmetric.
    s1 += expf(pl * LAMf + la);
    s2 += expf(lb - pl * LAMf);
  }
  for (int off = 16; off > 0; off >>= 1) {
    s1 += __shfl_down(s1, off, 32);
    s2 += __shfl_down(s2, off, 32);
  }
  if ((threadIdx.x & 31) == 0) {
    atomicAdd(&acc[0], (double)s1);
    atomicAdd(&acc[1], (double)s2);
  }
}

// ---------------------------------------------------------------------------
// K2: four-step stage 1. grid (256, 2): one 512-pt column FFT per block.
// j = n1 + 256*n2; FFT512 over n2 (two WMMA FFT256 + radix-2), then outer
// twiddle w_N^{n1*k2}; write A'[n1*512 + k2] to scratch (float2).
// ---------------------------------------------------------------------------
__global__ void k2_fft_cols(const float* __restrict__ histA,
                            const float* __restrict__ histB,
                            float2* __restrict__ FA, float2* __restrict__ FB) {
  __shared__ float sh[12 * 256];
  float* inRe = sh;            // [2][256] even/odd column samples
  float* inIm = sh + 512;
  float* midRe = sh + 1024;    // [2][256] per-wave matmul scratch
  float* midIm = sh + 1536;
  float* outRe = sh + 2048;    // [2][256] E (wave0) and O (wave1)
  float* outIm = sh + 2560;

  const int n1 = blockIdx.x;
  const float* hist = blockIdx.y ? histB : histA;
  float2* F = blockIdx.y ? FB : FA;
  const int tid = threadIdx.x;

  for (int e = tid; e < 512; e += 64) {
    const float v = hist[n1 + 256 * e];
    const int p = e & 1, m = e >> 1;  // decimation in time over n2
    inRe[p * 256 + m] = v;
    inIm[p * 256 + m] = 0.0f;
  }
  __syncthreads();

  const int w = tid >> 5, lane = tid & 31;
  fft256_wave(inRe + w * 256, inIm + w * 256, midRe + w * 256, midIm + w * 256,
              outRe + w * 256, outIm + w * 256, lane);

  for (int k = tid; k < 256; k += 64) {
    const float Er = outRe[k], Ei = outIm[k];
    const float Or = outRe[256 + k], Oi = outIm[256 + k];
    float s, c;
    sincosf(TWO_PI_F * (float)k * (1.0f / 512.0f), &s, &c);  // w512^k
    const float tr = Or * c + Oi * s;
    const float ti = Oi * c - Or * s;
    float xr[2], xi[2];
    xr[0] = Er + tr; xi[0] = Ei + ti;  // k2 = k
    xr[1] = Er - tr; xi[1] = Ei - ti;  // k2 = k + 256
#pragma unroll
    for (int hh = 0; hh < 2; ++hh) {
      const int k2 = k + 256 * hh;
      const int m = (n1 * k2) & (FFTN - 1);
      float s2v, c2v;
      sincosf(TWO_PI_F * (float)m * (1.0f / (float)FFTN), &s2v, &c2v);
      float2 o;
      o.x = xr[hh] * c2v + xi[hh] * s2v;  // * w_N^{n1 k2}
      o.y = xi[hh] * c2v - xr[hh] * s2v;
      F[n1 * 512 + k2] = o;
    }
  }
}

// ---------------------------------------------------------------------------
// K3: stage 2 + tail reduction. grid 1024 x 32 (1 wave / block).
// P[k2 + 512*k1] = FFT256_{n1}(A'[n1][k2]); accumulate
//   Re[ P^4 * (-1)^f * conj(W_f) ]  with analytic W_f (two geometric series).
// ---------------------------------------------------------------------------
__global__ void k3_fft_rows_reduce(const float2* __restrict__ FA,
                                   const float2* __restrict__ FB,
                                   double* __restrict__ acc, int minIndex,
                                   double r, double ra, double rN, double Cc) {
  __shared__ float sh[6 * 256];
  float* inRe = sh;        float* inIm = sh + 256;
  float* midRe = sh + 512; float* midIm = sh + 768;
  float* outRe = sh + 1024; float* outIm = sh + 1280;

  const int which = blockIdx.x >> 9;
  const int k2 = blockIdx.x & 511;
  const float2* F = which ? FB : FA;
  const int lane = threadIdx.x;

  for (int n1 = lane; n1 < 256; n1 += 32) {
    const float2 v = F[n1 * 512 + k2];
    inRe[n1] = v.x;
    inIm[n1] = v.y;
  }
  __syncthreads();
  fft256_wave(inRe, inIm, midRe, midIm, outRe, outIm, lane);

  const int a = minIndex + 1;
  double local = 0.0;
#pragma unroll
  for (int it = 0; it < 8; ++it) {
    const int k1 = lane + 32 * it;
    const double Pr = (double)outRe[k1], Pi = (double)outIm[k1];
    const int f = k2 + 512 * k1;
    const double p2r = Pr * Pr - Pi * Pi, p2i = 2.0 * Pr * Pi;
    const double p4r = p2r * p2r - p2i * p2i, p4i = 2.0 * p2r * p2i;
    // z = e^{-2pi i f/N}; za = z^a (arg reduced mod N in 64-bit)
    double zr, zi, zar, zai;
    sincos(-TWO_PI_D * (double)f / (double)FFTN, &zi, &zr);
    const long long fm = ((long long)f * (long long)a) & (long long)(FFTN - 1);
    sincos(-TWO_PI_D * (double)fm / (double)FFTN, &zai, &zar);
    double S1r, S1i;
    if (f == 0) {
      S1r = (double)(FFTN - a); S1i = 0.0;
    } else {  // (z^a - 1)/(1 - z)
      const double nr = zar - 1.0, ni = zai;
      const double dr = 1.0 - zr, di = -zi;
      const double dn = dr * dr + di * di;
      S1r = (nr * dr + ni * di) / dn;
      S1i = (ni * dr - nr * di) / dn;
    }
    // S2 = (r^a z^a - r^N)/(1 - r z)
    const double n2r = ra * zar - rN, n2i = ra * zai;
    const double d2r = 1.0 - r * zr, d2i = -r * zi;
    const double dn2 = d2r * d2r + d2i * d2i;
    const double S2r = (n2r * d2r + n2i * d2i) / dn2;
    const double S2i = (n2i * d2r - n2r * d2i) / dn2;
    const double Wr = S1r - Cc * S2r, Wi = S1i - Cc * S2i;
    double term = p4r * Wr + p4i * Wi;  // Re[p4 * conj(W)]
    if (f & 1) term = -term;            // fftshift phases
    local += term;
  }
  for (int off = 16; off > 0; off >>= 1) local += __shfl_down(local, off, 32);
  if (lane == 0) atomicAdd(&acc[2 + which], local);
}

// ---------------------------------------------------------------------------
// K4: scalar finalize
// ---------------------------------------------------------------------------
__global__ void k4_finalize(const double* __restrict__ acc,
                            const float* __restrict__ distEv,
                            float* __restrict__ out, double EF, double invN) {
  const double S = acc[0], Sm = acc[1];  // exp(alpha_plus), exp(alpha_minus)
  const double S2 = S * S, S4 = S2 * S2, S5 = S4 * S;
  const double Sm2 = Sm * Sm, Sm4 = Sm2 * Sm2, Sm5 = Sm4 * Sm;
  const double errA =
      ((2.0 * S5 - S4 - S) / (S - 1.0) + (Sm5 - Sm) / (Sm - 1.0)) * EF;
  const double errB =
      ((2.0 * Sm5 - Sm4 - Sm) / (Sm - 1.0) + (S5 - S) / (S - 1.0)) * EF;
  const double de = (double)distEv[0];
  const double q = 1.0 - de, q2 = q * q;
  const double dec = 1.0 - q2 * q2;  // 1 - (1-de)^M, M=4
  const double dA = dec + acc[2] * invN + errA;
  const double dB = dec + acc[3] * invN + errB;
  out[0] = (float)dA;
  out[1] = (float)dB;
  out[2] = 0.0f;
  out[3] = 0.0f;
  out[4] = (float)dA;
}

// ---------------------------------------------------------------------------
extern "C" void kernel_launch(void* const* d_in, const int* in_sizes, int n_in,
                              void* d_out, int out_size, void* d_ws,
                              size_t ws_size, hipStream_t stream) {
  (void)n_in; (void)out_size; (void)ws_size;
  const int N = FFTN;
  const float* pA = (const float*)d_in[0];
  const float* pB = (const float*)d_in[1];
  const float* distEv = (const float*)d_in[2];
  float* out = (float*)d_out;

  float* histA = (float*)d_ws;
  float* histB = histA + N;
  double* acc = (double*)(histB + N);     // [Splus, Sminus, tailA, tailB]
  float2* FA = (float2*)(acc + 8);        // stage-1 spectra, N complex each
  float2* FB = FA + N;

  // scalar constants in double (graph-capture safe: pure host arithmetic)
  const double Ld = log(1.00002) * 131072.0;
  const double DXd = 2.0 * Ld / (double)N;
  const double LAMd = Ld / 2.0;
  const int minIndex = (int)floor((double)N * (Ld + 1.0) / (2.0 * Ld));
  const double EF = exp(-LAMd * Ld) / (1.0 - exp(-2.0 * LAMd * Ld));
  const int a = minIndex + 1;
  const double r = exp(-DXd);
  const double ra = exp(-DXd * (double)a);
  const double rN = exp(-2.0 * Ld);       // r^N
  const double Cc = exp(1.0 + Ld);        // e^{EPS + L}
  const int n = in_sizes[0];

  k0_zero<<<(2 * N + 255) / 256, 256, 0, stream>>>(histA, acc);
  k1_stream<<<4096, 256, 0, stream>>>(pA, pB, histA, histB, acc, (float)Ld,
                                      (float)DXd, (float)LAMd, n);
  k2_fft_cols<<<dim3(256, 2), 64, 0, stream>>>(histA, histB, FA, FB);
  k3_fft_rows_reduce<<<1024, 32, 0, stream>>>(FA, FB, acc, minIndex, r, ra, rN,
                                              Cc);
  k4_finalize<<<1, 1, 0, stream>>>(acc, distEv, out, EF, 1.0 / (double)N);
}